// SpatialEmbLoss2_4990751998184
// MI455X (gfx1250) — compile-verified
//
#include <hip/hip_runtime.h>
#include <hip/hip_bf16.h>

#define PP (512*512)
#define NID 15
#define NSEG 60
#define INV511 (1.0f/511.0f)

typedef __attribute__((ext_vector_type(2))) float v2f;
typedef __attribute__((ext_vector_type(8))) float v8f;

#define HAVE_WMMA_F32X4 __has_builtin(__builtin_amdgcn_wmma_f32_16x16x4_f32)

__device__ __forceinline__ float sigmoidf_(float x) { return 1.0f / (1.0f + expf(-x)); }

// ---------------------------------------------------------------------------
// Pass 1: per (b,id) sums of {1, x, y, sigma} and per-b background seed^2 sum.
// Block-local LDS bins -> deterministic per-block partial slots in d_ws.
// ---------------------------------------------------------------------------
__global__ void k_pass1(const float* __restrict__ pred, const int* __restrict__ inst,
                        const int* __restrict__ lab, float* __restrict__ p1p) {
  __shared__ float bins[61];
  int tid = threadIdx.x;
  if (tid < 61) bins[tid] = 0.0f;
  __syncthreads();
  int b = blockIdx.y;
  int p = blockIdx.x * 256 + tid;
  int iv = inst[(size_t)b * PP + p];
  int lv = lab[(size_t)b * PP + p];
  float sg = pred[((size_t)(b * 4) + 2) * PP + p];
  float s3 = pred[((size_t)(b * 4) + 3) * PP + p];
  if (iv >= 1) {
    float x = (float)(p & 511) * INV511;
    float y = (float)(p >> 9) * INV511;
    int o = (iv - 1) * 4;
    atomicAdd(&bins[o + 0], 1.0f);
    atomicAdd(&bins[o + 1], x);
    atomicAdd(&bins[o + 2], y);
    atomicAdd(&bins[o + 3], sg);
  }
  float sd = sigmoidf_(s3);
  float bgv = (lv == 0) ? sd * sd : 0.0f;
  for (int off = 16; off >= 1; off >>= 1) bgv += __shfl_xor(bgv, off);
  if ((tid & 31) == 0) atomicAdd(&bins[60], bgv);
  __syncthreads();
  if (tid < 61) p1p[((size_t)(b * 1024) + blockIdx.x) * 61 + tid] = bins[tid];
}

__global__ void k_reduce_p1(const float* __restrict__ p1p, float* __restrict__ stats,
                            float* __restrict__ seedbg) {
  int t = threadIdx.x;
  if (t >= 244) return;
  int b = t / 61, o = t % 61;
  float s = 0.0f;
  for (int k = 0; k < 1024; ++k) s += p1p[((size_t)(b * 1024) + k) * 61 + o];
  if (o < 60) stats[b * 60 + o] = s;   // stats[(b*15+id)*4 + f]
  else        seedbg[b] = s;
}

__global__ void k_finalize(const float* __restrict__ stats, float* __restrict__ deriv) {
  int i = threadIdx.x;
  if (i >= NSEG) return;
  const float* st = stats + i * 4;
  float cnt = st[0];
  float cs = fmaxf(cnt, 1.0f);
  float cx = st[1] / cs, cy = st[2] / cs, sm = st[3] / cs;
  deriv[i * 4 + 0] = cx;
  deriv[i * 4 + 1] = cy;
  deriv[i * 4 + 2] = expf(sm * 10.0f);   // s
  deriv[i * 4 + 3] = sm;                 // sig_mean
}

// ---------------------------------------------------------------------------
// Pass 2: dist(p,id) via WMMA distance GEMM; accumulate var/intra/seed per id;
// emit packed error keys (float bits, label in LSB) for the Lovasz sort.
// ---------------------------------------------------------------------------
__global__ void __launch_bounds__(256) k_pass2(const float* __restrict__ pred,
                                               const int* __restrict__ inst,
                                               const float* __restrict__ deriv,
                                               unsigned* __restrict__ err,
                                               float* __restrict__ p2p) {
  int tid = threadIdx.x;
  int lane = tid & 31;
  int wv = blockIdx.x * 8 + (tid >> 5);   // wave index within image, 0..511
  int b = blockIdx.y;

  // lanes 0..14 hold per-instance derived stats (id = lane); others zero.
  float cx = 0.0f, cy = 0.0f, sE = 0.0f, sm = 0.0f;
  if (lane < NID) {
    const float* dv = deriv + (b * NID + lane) * 4;
    cx = dv[0]; cy = dv[1]; sE = dv[2]; sm = dv[3];
  }
  float cn = cx * cx + cy * cy;
  int hi8 = (lane >= 16) ? 8 : 0;
  float cnR[8], sR[8], smR[8], cxR[8], cyR[8];
#pragma unroll
  for (int r = 0; r < 8; ++r) {
    int m = r + hi8;
    cnR[r] = __shfl(cn, m);
    sR[r]  = __shfl(sE, m);
    smR[r] = __shfl(sm, m);
    cxR[r] = __shfl(cx, m);
    cyR[r] = __shfl(cy, m);
  }
#if HAVE_WMMA_F32X4
  // A (16x4 f32): lanes 0-15 = rows M (ids), VGPR0=K0(cx), VGPR1=K1(cy); K2/K3 lanes zero.
  v2f Am;
  Am.x = (lane < 16) ? cx : 0.0f;
  Am.y = (lane < 16) ? cy : 0.0f;
#endif
  float accV[8], accI[8], accS[8];
#pragma unroll
  for (int r = 0; r < 8; ++r) { accV[r] = 0.0f; accI[r] = 0.0f; accS[r] = 0.0f; }

  const float* ch0 = pred + (size_t)(b * 4 + 0) * PP;
  const float* ch1 = pred + (size_t)(b * 4 + 1) * PP;
  const float* ch2 = pred + (size_t)(b * 4 + 2) * PP;
  const float* ch3 = pred + (size_t)(b * 4 + 3) * PP;
  const int*   ib  = inst + (size_t)b * PP;

  for (int t = 0; t < 16; ++t) {
    int base = (t * 512 + wv) * 32;
    int p = base + lane;
    float e0 = tanhf(ch0[p]) + (float)(p & 511) * INV511;
    float e1 = tanhf(ch1[p]) + (float)(p >> 9) * INV511;
    float sg = ch2[p];
    float sd = sigmoidf_(ch3[p]);
    int iv = ib[p];
#pragma unroll
    for (int half = 0; half < 2; ++half) {
      int srcL = (lane & 15) + half * 16;
      float bx  = __shfl(e0, srcL);   // pixel tileBase+(lane&15) embedding, in all lanes
      float by  = __shfl(e1, srcL);
      float sgS = __shfl(sg, srcL);
      float sdS = __shfl(sd, srcL);
      int   ivS = __shfl(iv, srcL);
      float pn = bx * bx + by * by;
      int pix = base + half * 16 + (lane & 15);
#if HAVE_WMMA_F32X4
      // B (4x16 f32): lanes 0-15 = cols N (pixels), VGPR0=K0(ex), VGPR1=K1(ey).
      v2f Bm;
      Bm.x = (lane < 16) ? bx : 0.0f;
      Bm.y = (lane < 16) ? by : 0.0f;
      v8f c = {};
      v8f d = __builtin_amdgcn_wmma_f32_16x16x4_f32(
          false, Am, false, Bm, (short)0, c, false, false);
#endif
#pragma unroll
      for (int r = 0; r < 8; ++r) {
        int m = r + hi8;                  // id row of D
#if HAVE_WMMA_F32X4
        float dot = d[r];                 // <c(m), e(pixel)>
#else
        float dot = cxR[r] * bx + cyR[r] * by;
#endif
        float d2 = pn + cnR[r] - 2.0f * dot;
        float dist = expf(-d2 * sR[r]);
        bool mf = (ivS == m + 1);
        float mff = mf ? 1.0f : 0.0f;
        float dv = sgS - smR[r];
        accV[r] += mff * dv * dv;
        accI[r] += mff * dist;
        float sv = sdS - dist;
        accS[r] += mff * sv * sv;
        float e = mf ? (2.0f - 2.0f * dist) : (2.0f * dist);   // errors >= 0
        if (m < NID) {
          unsigned u = (__float_as_uint(e) & ~1u) | (mf ? 1u : 0u);
          err[(size_t)(b * NID + m) * PP + pix] = u;
        }
      }
    }
  }
  // Deterministic per-wave partial write-out (one slot per (b,id,wave)).
#pragma unroll
  for (int r = 0; r < 8; ++r) {
    float v0 = accV[r], v1 = accI[r], v2 = accS[r];
    for (int off = 8; off >= 1; off >>= 1) {
      v0 += __shfl_xor(v0, off);
      v1 += __shfl_xor(v1, off);
      v2 += __shfl_xor(v2, off);
    }
    int m = r + hi8;
    if ((lane & 15) == 0 && m < NID) {
      float* dst = p2p + ((size_t)(b * NID + m) * 512 + wv) * 3;
      dst[0] = v0; dst[1] = v1; dst[2] = v2;
    }
  }
}

__global__ void k_reduce_p2(const float* __restrict__ p2p, float* __restrict__ acc) {
  int t = threadIdx.x;
  if (t >= 180) return;
  int b = t / 45, rem = t % 45, id = rem / 3, f = rem % 3;
  float s = 0.0f;
  for (int w = 0; w < 512; ++w) s += p2p[((size_t)(b * NID + id) * 512 + w) * 3 + f];
  acc[(b * NID + id) * 4 + f] = s;   // 0=var_sum, 1=intra_sum, 2=seed_sum
}

// ---------------------------------------------------------------------------
// Descending bitonic sort of 60 independent 262144-element u32 segments.
// Local kernel: all passes with j <= 2048 inside one 4096-element LDS block.
// ---------------------------------------------------------------------------
__global__ void __launch_bounds__(1024) k_bitonic_local(unsigned* __restrict__ a,
                                                        int kFirst, int kLast) {
  __shared__ unsigned s[4096];
  int seg = blockIdx.x >> 6;
  int chunk = blockIdx.x & 63;
  size_t base = (size_t)seg * PP + (size_t)chunk * 4096;
  int gbase = chunk * 4096;          // within-segment offset of this block
  int tid = threadIdx.x;
  s[tid]        = a[base + tid];
  s[tid + 1024] = a[base + tid + 1024];
  s[tid + 2048] = a[base + tid + 2048];
  s[tid + 3072] = a[base + tid + 3072];
  __syncthreads();
  for (int k = kFirst; k <= kLast; k <<= 1) {
    int jStart = ((k >> 1) < 2048) ? (k >> 1) : 2048;
    for (int j = jStart; j >= 1; j >>= 1) {
      for (int c = tid; c < 2048; c += 1024) {
        int i = ((c & ~(j - 1)) << 1) | (c & (j - 1));
        int l = i | j;
        bool desc = (((gbase + i) & k) == 0);
        unsigned x = s[i], y = s[l];
        if (desc ? (x < y) : (x > y)) { s[i] = y; s[l] = x; }
      }
      __syncthreads();
    }
  }
  a[base + tid]        = s[tid];
  a[base + tid + 1024] = s[tid + 1024];
  a[base + tid + 2048] = s[tid + 2048];
  a[base + tid + 3072] = s[tid + 3072];
}

__global__ void k_bitonic_global(unsigned* __restrict__ a, int k, int j) {
  int c = blockIdx.x * 256 + threadIdx.x;   // 60 * 131072 comparators
  int seg = c >> 17;
  int cs = c & 131071;
  int i = ((cs & ~(j - 1)) << 1) | (cs & (j - 1));
  int l = i | j;
  bool desc = ((i & k) == 0);
  size_t sb = (size_t)seg * PP;
  unsigned x = a[sb + i], y = a[sb + l];
  if (desc ? (x < y) : (x > y)) { a[sb + i] = y; a[sb + l] = x; }
}

// ---------------------------------------------------------------------------
// Lovasz hinge over each sorted segment: running positive-count scan + sum of
// e_i * (jac_i - jac_{i-1}).  One block per segment, 32 waves.
// ---------------------------------------------------------------------------
__global__ void __launch_bounds__(1024) k_lovasz(const unsigned* __restrict__ err,
                                                 const float* __restrict__ stats,
                                                 float* __restrict__ instl) {
  __shared__ int wsum[32];
  __shared__ int carrySh;
  __shared__ float red[32];
  int seg = blockIdx.x;
  int tid = threadIdx.x;
  int laneId = tid & 31, waveId = tid >> 5;
  float G = stats[seg * 4 + 0];
  if (tid == 0) carrySh = 0;
  __syncthreads();
  const unsigned* a = err + (size_t)seg * PP;
  float accL = 0.0f;
  for (int cb = 0; cb < PP; cb += 1024) {
    unsigned u = a[cb + tid];
    int g = (int)(u & 1u);
    float e = __uint_as_float(u & ~1u);
    int v = g;
    for (int off = 1; off < 32; off <<= 1) {
      int t2 = __shfl_up(v, off);
      if (laneId >= off) v += t2;
    }
    if (laneId == 31) wsum[waveId] = v;
    __syncthreads();
    if (waveId == 0) {
      int sv = wsum[laneId];
      for (int off = 1; off < 32; off <<= 1) {
        int t2 = __shfl_up(sv, off);
        if (laneId >= off) sv += t2;
      }
      wsum[laneId] = sv;
    }
    __syncthreads();
    int carry = carrySh;
    int waveOff = (waveId == 0) ? 0 : wsum[waveId - 1];
    int Pc = carry + waveOff + v;          // inclusive positive count at rank i
    int i = cb + tid;
    float If = (float)(i + 1);
    float Pcf = (float)Pc;
    float jac = 1.0f - (G - Pcf) / (G + If - Pcf);
    float jprev = 0.0f;
    if (i > 0) {
      float Pp = Pcf - (float)g;
      jprev = 1.0f - (G - Pp) / (G + (If - 1.0f) - Pp);
    }
    accL += e * (jac - jprev);
    __syncthreads();
    if (tid == 0) carrySh = carry + wsum[31];
    __syncthreads();
  }
  for (int off = 16; off >= 1; off >>= 1) accL += __shfl_xor(accL, off);
  if (laneId == 0) red[waveId] = accL;
  __syncthreads();
  if (tid == 0) {
    float s = 0.0f;
    for (int kk = 0; kk < 32; ++kk) s += red[kk];
    instl[seg] = s;
  }
}

__global__ void k_final(const float* __restrict__ stats, const float* __restrict__ seedbg,
                        const float* __restrict__ acc, const float* __restrict__ instl,
                        float* __restrict__ out) {
  __shared__ float lb[4];
  int t = threadIdx.x;
  if (t < 4) {
    int b = t;
    float obj = 0.0f, iL = 0.0f, vL = 0.0f, sFg = 0.0f, inL = 0.0f;
    for (int id = 0; id < NID; ++id) {
      int i = b * NID + id;
      float cnt = stats[i * 4];
      if (cnt > 0.0f) {
        obj += 1.0f;
        iL += instl[i];
        vL += acc[i * 4 + 0] / cnt;
        inL += 1.0f - acc[i * 4 + 1] / cnt;
        sFg += acc[i * 4 + 2];
      }
    }
    float objs = fmaxf(obj, 1.0f);
    float seed_loss = (seedbg[b] + 1.0f * sFg) / (float)PP;
    lb[b] = 1.0f * (iL / objs) + 10.0f * (vL / objs) + 1.0f * seed_loss + 5.0f * inL;
  }
  __syncthreads();
  if (t == 0) out[0] = 0.25f * (lb[0] + lb[1] + lb[2] + lb[3]);
}

extern "C" void kernel_launch(void* const* d_in, const int* in_sizes, int n_in,
                              void* d_out, int out_size, void* d_ws, size_t ws_size,
                              hipStream_t stream) {
  (void)in_sizes; (void)n_in; (void)out_size; (void)ws_size;
  const float* pred = (const float*)d_in[0];
  const int*   inst = (const int*)d_in[1];
  const int*   lab  = (const int*)d_in[2];
  float* out = (float*)d_out;

  float* W      = (float*)d_ws;
  float* P1P    = W;                       // 4*1024*61 = 249856 floats
  float* STATS  = W + 262144;              // 60*4
  float* SEEDBG = W + 262144 + 240;        // 4
  float* DERIV  = W + 262144 + 256;        // 60*4
  float* P2P    = W + 262656;              // 4*15*512*3 = 92160
  float* ACC    = W + 354816;              // 60*4
  float* INSTL  = W + 355072;              // 60
  unsigned* ERR = (unsigned*)(W + 524288); // 60*262144 u32 = 63 MB

  k_pass1<<<dim3(1024, 4), 256, 0, stream>>>(pred, inst, lab, P1P);
  k_reduce_p1<<<1, 256, 0, stream>>>(P1P, STATS, SEEDBG);
  k_finalize<<<1, 64, 0, stream>>>(STATS, DERIV);
  k_pass2<<<dim3(64, 4), 256, 0, stream>>>(pred, inst, DERIV, ERR, P2P);
  k_reduce_p2<<<1, 256, 0, stream>>>(P2P, ACC);

  // Descending bitonic sort of all 60 segments (N = 262144 each).
  k_bitonic_local<<<3840, 1024, 0, stream>>>(ERR, 2, 4096);        // stages k=2..4096
  for (int k = 8192; k <= 262144; k <<= 1) {
    for (int j = k >> 1; j >= 4096; j >>= 1)
      k_bitonic_global<<<30720, 256, 0, stream>>>(ERR, k, j);
    k_bitonic_local<<<3840, 1024, 0, stream>>>(ERR, k, k);         // j <= 2048 in LDS
  }

  k_lovasz<<<60, 1024, 0, stream>>>(ERR, STATS, INSTL);
  k_final<<<1, 32, 0, stream>>>(STATS, SEEDBG, ACC, INSTL, out);
}